// HashDecoder_56152402428242
// MI455X (gfx1250) — compile-verified
//
#include <hip/hip_runtime.h>

#define NLV 16            // levels
#define TBL (1u << 19)    // hash table size per level (power of two)
#define TPB 256           // threads per block (8 waves of 32)
#define PPB 128           // points per block
#define STR 34            // LDS row stride in floats (even + conflict-free)

typedef __attribute__((ext_vector_type(2))) float v2f;
typedef __attribute__((ext_vector_type(8))) float v8f;

__global__ __launch_bounds__(TPB) void hash_mlp_kernel(
    const float* __restrict__ p,    // [N,3]
    const float* __restrict__ emb,  // [16, 2^19, 2]
    const float* __restrict__ W1,   // [32,32]
    const float* __restrict__ W2,   // [32,32]
    const float* __restrict__ W3,   // [32,4]
    float* __restrict__ out)        // [N,4]
{
    __shared__ float s_enc[PPB * STR];      // encoded features [128][32] padded
    __shared__ float s_h[8 * 16 * STR];     // per-wave hidden tile [16][32] padded
    __shared__ float s_w1[32 * STR];        // W1 transposed: [n][k]
    __shared__ float s_w2[32 * STR];        // W2 transposed: [n][k]
    __shared__ float s_w3[16 * STR];        // W3 transposed + zero-padded to n=16

    const int t = threadIdx.x;

    // ---- cooperative weight staging (transposed so B-fragments are contiguous) ----
    for (int i = t; i < 1024; i += TPB) {
        const int k = i >> 5, n = i & 31;
        s_w1[n * STR + k] = W1[i];
        s_w2[n * STR + k] = W2[i];
    }
    for (int i = t; i < 512; i += TPB) {
        const int n = i >> 5, k = i & 31;
        s_w3[n * STR + k] = (n < 4) ? W3[k * 4 + n] : 0.0f;
    }

    // ---- hash-grid encoding: thread = (point row, level) ----
    const int level = t & 15;
    const int prow  = t >> 4;                  // 0..15
    const int pbase = blockIdx.x * PPB;

    // SCALES[l] = 16 * 2^(0.4*l) - 1 ; RES = ceil(scale)+1 ; hash iff RES^3 > T
    const float scale  = (float)(16.0 * exp2(0.4 * (double)level) - 1.0);
    const int   res    = (int)ceilf(scale) + 1;
    const bool  use_hash = ((long long)res * res * res) > (long long)TBL;
    const float* embL  = emb + (size_t)level * TBL * 2u;

    #pragma unroll
    for (int kk = 0; kk < PPB / 16; ++kk) {
        const int pt = kk * 16 + prow;         // local point 0..127
        const int gp = pbase + pt;
        float xs[3];
        xs[0] = p[gp * 3 + 0];
        xs[1] = p[gp * 3 + 1];
        xs[2] = p[gp * 3 + 2];

        int   ig[3];
        float w[3];
        #pragma unroll
        for (int d = 0; d < 3; ++d) {
            const float pos = xs[d] * scale + 0.5f;
            const float fl  = floorf(pos);
            const float fr  = pos - fl;
            ig[d] = (int)fl;
            w[d]  = fr * fr * (3.0f - 2.0f * fr);   // smoothstep weight
        }

        float a0 = 0.0f, a1 = 0.0f;
        #pragma unroll
        for (int c = 0; c < 8; ++c) {
            const int bx = c & 1, by = (c >> 1) & 1, bz = (c >> 2) & 1;
            unsigned idx;
            if (use_hash) {
                idx = ((unsigned)(ig[0] + bx)
                     ^ ((unsigned)(ig[1] + by) * 2654435761u
                     ^  (unsigned)(ig[2] + bz) * 805459861u)) & (TBL - 1u);
            } else {
                idx = (unsigned)((ig[0] + bx) + (ig[1] + by) * res
                                 + (ig[2] + bz) * res * res);
            }
            const v2f f = *(const v2f*)(embL + (size_t)idx * 2u);  // L2-resident gather
            const float cw = (bx ? w[0] : 1.0f - w[0])
                           * (by ? w[1] : 1.0f - w[1])
                           * (bz ? w[2] : 1.0f - w[2]);
            a0 += f.x * cw;
            a1 += f.y * cw;
        }
        v2f ev; ev.x = a0; ev.y = a1;
        *(v2f*)&s_enc[pt * STR + level * 2] = ev;
    }
    __syncthreads();

    // ---- MLP via V_WMMA_F32_16X16X4_F32 : one 16-point tile per wave ----
    const int wave  = t >> 5;
    const int lane  = t & 31;
    const int rbase = wave * 16;
    float* hw = s_h + wave * (16 * STR);

    const int mrow = lane & 15;         // A row / B col / D col
    const int koff = (lane >> 4) * 2;   // K sub-offset for A/B fragments
    const int crow = (lane >> 4) * 8;   // D row offset

    // GEMM1: h = relu(enc @ W1)
    v8f acc0 = {0,0,0,0,0,0,0,0};
    v8f acc1 = {0,0,0,0,0,0,0,0};
    #pragma unroll
    for (int k0 = 0; k0 < 32; k0 += 4) {
        const v2f a  = *(const v2f*)&s_enc[(rbase + mrow) * STR + k0 + koff];
        const v2f b0 = *(const v2f*)&s_w1[mrow * STR + k0 + koff];
        const v2f b1 = *(const v2f*)&s_w1[(16 + mrow) * STR + k0 + koff];
        acc0 = __builtin_amdgcn_wmma_f32_16x16x4_f32(false, a, false, b0, (short)0, acc0, false, false);
        acc1 = __builtin_amdgcn_wmma_f32_16x16x4_f32(false, a, false, b1, (short)0, acc1, false, false);
    }
    #pragma unroll
    for (int j = 0; j < 8; ++j) {
        hw[(crow + j) * STR + mrow]      = fmaxf(acc0[j], 0.0f);
        hw[(crow + j) * STR + 16 + mrow] = fmaxf(acc1[j], 0.0f);
    }

    // GEMM2: h = relu(h @ W2)  (reads complete before overwrite; wave-private tile)
    v8f acc2 = {0,0,0,0,0,0,0,0};
    v8f acc3 = {0,0,0,0,0,0,0,0};
    #pragma unroll
    for (int k0 = 0; k0 < 32; k0 += 4) {
        const v2f a  = *(const v2f*)&hw[mrow * STR + k0 + koff];
        const v2f b0 = *(const v2f*)&s_w2[mrow * STR + k0 + koff];
        const v2f b1 = *(const v2f*)&s_w2[(16 + mrow) * STR + k0 + koff];
        acc2 = __builtin_amdgcn_wmma_f32_16x16x4_f32(false, a, false, b0, (short)0, acc2, false, false);
        acc3 = __builtin_amdgcn_wmma_f32_16x16x4_f32(false, a, false, b1, (short)0, acc3, false, false);
    }
    #pragma unroll
    for (int j = 0; j < 8; ++j) {
        hw[(crow + j) * STR + mrow]      = fmaxf(acc2[j], 0.0f);
        hw[(crow + j) * STR + 16 + mrow] = fmaxf(acc3[j], 0.0f);
    }

    // GEMM3: out = h @ W3 (padded to 16 cols; only first 4 valid)
    v8f acc4 = {0,0,0,0,0,0,0,0};
    #pragma unroll
    for (int k0 = 0; k0 < 32; k0 += 4) {
        const v2f a = *(const v2f*)&hw[mrow * STR + k0 + koff];
        const v2f b = *(const v2f*)&s_w3[mrow * STR + k0 + koff];
        acc4 = __builtin_amdgcn_wmma_f32_16x16x4_f32(false, a, false, b, (short)0, acc4, false, false);
    }
    if (mrow < 4) {
        #pragma unroll
        for (int j = 0; j < 8; ++j) {
            const int gp = pbase + rbase + crow + j;
            out[gp * 4 + mrow] = acc4[j];
        }
    }
}

extern "C" void kernel_launch(void* const* d_in, const int* in_sizes, int n_in,
                              void* d_out, int out_size, void* d_ws, size_t ws_size,
                              hipStream_t stream) {
    const float* p  = (const float*)d_in[0];
    const float* e  = (const float*)d_in[1];
    const float* W1 = (const float*)d_in[2];
    const float* W2 = (const float*)d_in[3];
    const float* W3 = (const float*)d_in[4];
    float* out = (float*)d_out;

    const int npts   = in_sizes[0] / 3;      // 524288
    const int blocks = npts / PPB;           // 4096 (N is a power of two)
    hipLaunchKernelGGL(hash_mlp_kernel, dim3(blocks), dim3(TPB), 0, stream,
                       p, e, W1, W2, W3, out);
}